// BoundingBox_LossProcessor_14156212208285
// MI455X (gfx1250) — compile-verified
//
#include <hip/hip_runtime.h>
#include <hip/hip_bf16.h>

#define NBOX 8192
#define NCLS 20
#define KTOP 20
#define CONF_T 0.6f
#define IOU_T 0.5f

typedef __attribute__((ext_vector_type(2))) float v2f;
typedef __attribute__((ext_vector_type(8))) float v8f;

// ---------------------------------------------------------------- helpers
__device__ __forceinline__ float iou_f(float4 a, float aa, float4 b, float ab) {
    float xx1 = fmaxf(a.x, b.x), yy1 = fmaxf(a.y, b.y);
    float xx2 = fminf(a.z, b.z), yy2 = fminf(a.w, b.w);
    float w = fmaxf(xx2 - xx1, 0.f), h = fmaxf(yy2 - yy1, 0.f);
    float inter = w * h;
    return inter / (aa + ab - inter);
}

// ---------------------------------------------------------------- K1: filter
// scores = max over classes; keep boxes with score > CONF_T; ordered compaction
__global__ void k_filter(const float* __restrict__ conf, int* __restrict__ filt,
                         int* __restrict__ hdr) {
    __shared__ int ssum[1024];
    int tid = threadIdx.x;
    int flags[8];
    int local = 0;
    for (int u = 0; u < 8; ++u) {
        int b = tid * 8 + u;
        const float* cp = conf + b * NCLS;
        float m = cp[0];
        for (int c = 1; c < NCLS; ++c) m = fmaxf(m, cp[c]);
        flags[u] = (m > CONF_T) ? 1 : 0;
        local += flags[u];
    }
    ssum[tid] = local;
    __syncthreads();
    // inclusive Hillis-Steele scan over 1024 per-thread sums
    for (int off = 1; off < 1024; off <<= 1) {
        int v = (tid >= off) ? ssum[tid - off] : 0;
        __syncthreads();
        ssum[tid] += v;
        __syncthreads();
    }
    int pos = ssum[tid] - local;  // exclusive
    for (int u = 0; u < 8; ++u) {
        if (flags[u]) filt[pos++] = tid * 8 + u;
    }
    if (tid == 1023) hdr[0] = ssum[1023];
}

// ---------------------------------------------------------------- K2: sort
// stable descending sort by nms_score = conf[filt_idx, 0]; 64KB LDS bitonic
__global__ void k_sort(const float* __restrict__ loc, const float* __restrict__ conf,
                       const int* __restrict__ filt, const int* __restrict__ hdr,
                       int* __restrict__ sorig, float4* __restrict__ sbox,
                       float* __restrict__ sarea) {
    __shared__ unsigned long long key[NBOX];  // 64 KB (CDNA5: 320 KB/WGP)
    int tid = threadIdx.x;
    int Mf = hdr[0];
    for (int i = tid; i < NBOX; i += 1024) {
        unsigned long long kk = 0ull;
        if (i < Mf) {
            unsigned fb = __float_as_uint(conf[filt[i] * NCLS + 0]);
            kk = ((unsigned long long)fb << 32) | (unsigned)(0xFFFFFFFFu - (unsigned)i);
        }
        key[i] = kk;
    }
    __syncthreads();
    for (int k = 2; k <= NBOX; k <<= 1) {
        for (int j = k >> 1; j > 0; j >>= 1) {
            for (int i = tid; i < NBOX; i += 1024) {
                int ixj = i ^ j;
                if (ixj > i) {
                    unsigned long long a = key[i], b = key[ixj];
                    bool dir = ((i & k) == 0);  // dir=true -> descending run
                    if ((a < b) == dir) { key[i] = b; key[ixj] = a; }
                }
            }
            __syncthreads();
        }
    }
    for (int i = tid; i < NBOX; i += 1024) {
        if (i < Mf) {
            unsigned pos = 0xFFFFFFFFu - (unsigned)(key[i] & 0xFFFFFFFFull);
            int oi = filt[pos];
            sorig[i] = oi;
            float4 bx = ((const float4*)loc)[oi];
            sbox[i] = bx;
            sarea[i] = (bx.z - bx.x) * (bx.w - bx.y);
        }
    }
}

// ---------------------------------------------------------------- K3: NMS
// chunked greedy NMS: wave32 ballot resolves intra-chunk dependencies,
// whole block applies suppression to later boxes. Chunk boxes are staged
// to LDS with the CDNA5 async global->LDS path.
__global__ void k_nms(const float4* __restrict__ sbox, const float* __restrict__ sarea,
                      const int* __restrict__ hdr, const int* __restrict__ sorig,
                      int* __restrict__ keepo, int* __restrict__ nkeep) {
    __shared__ unsigned removed[NBOX / 32];
    __shared__ unsigned keepmask[NBOX / 32];
    __shared__ float4 cb[32];
    __shared__ float ca[32];
    __shared__ int cpre[257];
    int tid = threadIdx.x;
    int Mf = hdr[0];
    for (int i = tid; i < NBOX / 32; i += 1024) { removed[i] = 0u; keepmask[i] = 0u; }
    __syncthreads();

    int nchunk = (Mf + 31) / 32;
    for (int ch = 0; ch < nchunk; ++ch) {
        int base = ch * 32;
        if (tid < 32) {
            // async copy 32 boxes (512B) into LDS (gfx1250 async-to-LDS path)
            unsigned lds_off = (unsigned)(size_t)(void*)&cb[tid];
            unsigned long long gaddr =
                (unsigned long long)(size_t)(const void*)(sbox + base + tid);
            asm volatile("global_load_async_to_lds_b128 %0, %1, off"
                         :: "v"(lds_off), "v"(gaddr) : "memory");
            asm volatile("s_wait_asynccnt 0" ::: "memory");
            ca[tid] = sarea[base + tid];
        }
        __syncthreads();
        if (tid < 32) {  // wave 0: intra-chunk greedy via ballot
            int j = tid;
            float4 bj = cb[j];
            float aj = ca[j];
            unsigned m = 0u;  // bit i: box i (i<j) suppresses me
            for (int i2 = 0; i2 < 32; ++i2) {
                if (i2 < j) {
                    if (iou_f(cb[i2], ca[i2], bj, aj) > IOU_T) m |= (1u << i2);
                }
            }
            unsigned alive = ~removed[ch];
            int rem = Mf - base;
            if (rem < 32) alive &= (1u << rem) - 1u;
            unsigned kept = 0u;
            for (int i2 = 0; i2 < 32; ++i2) {
                unsigned supp = (unsigned)__ballot((m >> i2) & 1u);
                if ((alive >> i2) & 1u) { kept |= (1u << i2); alive &= ~supp; }
            }
            if (j == 0) keepmask[ch] = kept;
        }
        __syncthreads();
        unsigned kept = keepmask[ch];
        // suppress all later boxes against this chunk's kept set
        for (int jj = base + 32 + tid; jj < Mf; jj += 1024) {
            unsigned w = removed[jj >> 5];
            if ((w >> (jj & 31)) & 1u) continue;
            float4 bj = sbox[jj];
            float aj = sarea[jj];
            unsigned km = kept;
            bool sup = false;
            while (km) {
                int i2 = __ffs(km) - 1;
                km &= km - 1;
                if (iou_f(cb[i2], ca[i2], bj, aj) > IOU_T) { sup = true; break; }
            }
            if (sup) atomicOr(&removed[jj >> 5], 1u << (jj & 31));
        }
        __syncthreads();
    }
    // build ordered keep list
    if (tid == 0) {
        int acc = 0;
        for (int c = 0; c < NBOX / 32; ++c) { cpre[c] = acc; acc += __popc(keepmask[c]); }
        cpre[NBOX / 32] = acc;
        *nkeep = acc;
    }
    __syncthreads();
    if (tid < NBOX / 32) {
        unsigned km = keepmask[tid];
        int p = cpre[tid];
        int b2 = tid * 32;
        while (km) {
            int b = __ffs(km) - 1;
            km &= km - 1;
            keepo[p++] = sorig[b2 + b];
        }
    }
}

// ---------------------------------------------------------------- K4: top-k
// per-class top-20 over kept boxes (value desc, index asc on ties),
// selection-by-lexicographic-bound, 64-bit LDS atomicMax reduction.
__global__ void k_topk(const float* __restrict__ conf, const int* __restrict__ keepo,
                       const int* __restrict__ nkeep, float* __restrict__ topv,
                       int* __restrict__ topi) {
    __shared__ unsigned long long best;
    int c = blockIdx.x;
    int tid = threadIdx.x;
    int M = *nkeep;
    float prev_v = 1e30f;
    int prev_i = -1;
    for (int r = 0; r < KTOP; ++r) {
        if (tid == 0) best = 0ull;
        __syncthreads();
        unsigned long long lb = 0ull;
        for (int m = tid; m < M; m += blockDim.x) {
            float v = conf[keepo[m] * NCLS + c];
            if (v < prev_v || (v == prev_v && m > prev_i)) {
                unsigned long long kk = ((unsigned long long)__float_as_uint(v) << 32) |
                                        (unsigned)(0x7FFFFFFF - m);
                if (kk > lb) lb = kk;
            }
        }
        atomicMax(&best, lb);
        __syncthreads();
        unsigned long long b = best;
        float bv;
        int bi;
        if (b == 0ull) { bv = 0.0f; bi = 0; }
        else {
            bv = __uint_as_float((unsigned)(b >> 32));
            bi = 0x7FFFFFFF - (int)(unsigned)(b & 0xFFFFFFFFull);
        }
        prev_v = bv;
        prev_i = bi;
        if (tid == 0) { topv[r * NCLS + c] = bv; topi[r * NCLS + c] = bi; }
        __syncthreads();
    }
}

// ---------------------------------------------------------------- K5: loss
__device__ __forceinline__ float sl1_term(int flat, const float* loc, const float* tb,
                                          const int* ko, const int* ti) {
    int r = flat / 80;
    int rem = flat - r * 80;
    int c = rem >> 2;
    int d = rem & 3;
    int m = ti[r * NCLS + c];
    float pred = loc[ko[m] * 4 + d];
    float dd = fabsf(pred - tb[c * 4 + d]);
    return (dd < 1.f) ? 0.5f * dd * dd : dd - 0.5f;
}

// one wave (32 threads). 1600 smooth-L1 terms reduced on the matrix core:
// 25 x V_WMMA_F32_16X16X4_F32 with B = ones (exact f32 row-sum reduction).
__global__ void k_loss(const float* __restrict__ loc, const float* __restrict__ tb,
                       const int* __restrict__ tl, const int* __restrict__ hdr,
                       const int* __restrict__ keepo, const float* __restrict__ topv,
                       const int* __restrict__ topi, float* __restrict__ out) {
    int lane = threadIdx.x;
    int mf = hdr[0], nk = hdr[1];
    if (mf == 0 || nk == 0) {  // uniform branch
        if (lane == 0) out[0] = 0.001f;
        return;
    }
    int hf = lane >> 4;      // K-half (0: K=0,1  1: K=2,3)
    int mrow = lane & 15;    // M row
    v8f acc = {};
    v2f B;
    B[0] = 1.0f;
    B[1] = 1.0f;
    for (int ch = 0; ch < 25; ++ch) {
        int e0 = ch * 64 + mrow * 4 + hf * 2;  // flat = M*4 + K
        v2f A;
        A[0] = sl1_term(e0, loc, tb, keepo, topi);
        A[1] = sl1_term(e0 + 1, loc, tb, keepo, topi);
        acc = __builtin_amdgcn_wmma_f32_16x16x4_f32(
            false, A, false, B, (short)0, acc, false, false);
    }
    float s = acc[0] + acc[1] + acc[2] + acc[3] + acc[4] + acc[5] + acc[6] + acc[7];
    // lanes 0-15 hold sum of rows 0-7, lanes 16-31 rows 8-15 (all columns equal)
    float loc_loss = __shfl(s, 0, 32) + __shfl(s, 16, 32);
    if (lane == 0) {
        float cb[KTOP];
        for (int r = 0; r < KTOP; ++r) cb[r] = (topv[r * NCLS + 0] > 0.5f) ? 1.f : 0.f;
        float mx = 0.f;
        for (int r = 0; r < KTOP; ++r) mx = fmaxf(mx, cb[r]);
        float se = 0.f;
        for (int r = 0; r < KTOP; ++r) se += expf(cb[r] - mx);
        float lse = mx + logf(se);
        float ce = 0.f;
        for (int r = 0; r < KTOP; ++r) ce -= (float)tl[r] * (cb[r] - lse);
        float pt = expf(-ce);
        float closs = 0.25f * (1.f - pt) * (1.f - pt) * ce;
        out[0] = (loc_loss + closs) / (float)nk;
    }
}

// ---------------------------------------------------------------- launch
extern "C" void kernel_launch(void* const* d_in, const int* in_sizes, int n_in,
                              void* d_out, int out_size, void* d_ws, size_t ws_size,
                              hipStream_t stream) {
    const float* loc = (const float*)d_in[0];    // (1, 8192, 4)
    const float* conf = (const float*)d_in[1];   // (1, 8192, 20)
    const float* tb = (const float*)d_in[2];     // (20, 4)
    const int* tl = (const int*)d_in[3];         // (20,)
    float* out = (float*)d_out;

    char* w = (char*)d_ws;
    int* hdr = (int*)w;                             // [0]=Mf, [1]=nkeep
    int* filt = (int*)(w + 64);                     // 8192 ints
    int* sorig = (int*)(w + 64 + 32768);            // 8192 ints
    float4* sbox = (float4*)(w + 65600);            // 8192 float4 (16B aligned)
    float* sarea = (float*)(w + 196672);            // 8192 floats
    int* keepo = (int*)(w + 229440);                // 8192 ints
    float* topv = (float*)(w + 262208);             // 400 floats
    int* topi = (int*)(w + 263808);                 // 400 ints

    k_filter<<<1, 1024, 0, stream>>>(conf, filt, hdr);
    k_sort<<<1, 1024, 0, stream>>>(loc, conf, filt, hdr, sorig, sbox, sarea);
    k_nms<<<1, 1024, 0, stream>>>(sbox, sarea, hdr, sorig, keepo, hdr + 1);
    k_topk<<<NCLS, 256, 0, stream>>>(conf, keepo, hdr + 1, topv, topi);
    k_loss<<<1, 32, 0, stream>>>(loc, tb, tl, hdr, keepo, topv, topi, out);
    (void)in_sizes; (void)n_in; (void)out_size; (void)ws_size;
}